// matcher_78262894068289
// MI455X (gfx1250) — compile-verified
//
#include <hip/hip_runtime.h>
#include <cmath>

typedef __attribute__((ext_vector_type(2))) float v2f;
typedef __attribute__((ext_vector_type(8))) float v8f;

namespace {

constexpr int Bdim = 2;
constexpr int Cdim = 128;
constexpr int Hn   = 4;
constexpr int HD   = 32;
constexpr int Ndim = 3000;   // compile-time: lets strides fold into load offsets
constexpr int Kn   = 64;
constexpr int NCP  = 3001;   // couplings dim (N + dustbin)
constexpr int UVS  = 3008;   // padded stride for u/v vectors
constexpr int SINK_ITERS = 100;

__device__ __forceinline__ v8f wmma4(v2f a, v2f b, v8f c) {
  // V_WMMA_F32_16X16X4_F32 : D(16x16,f32) = A(16x4,f32) * B(4x16,f32) + C
  return __builtin_amdgcn_wmma_f32_16x16x4_f32(
      /*neg_a=*/false, a, /*neg_b=*/false, b,
      /*c_mod=*/(short)0, c, /*reuse_a=*/false, /*reuse_b=*/false);
}

// Shared 16x64 WMMA inner step: one A fragment vs 4 B fragments.
// bp points at row (k0+half*2) of the B source; columns are colc[0..3].
// b.y uses +Ndim element offsets (a.y: +a_stride_y) -> immediate 12000-byte load offsets.
__device__ __forceinline__ void mma_step(const float* __restrict__ ap,
                                         const float* __restrict__ bp,
                                         const int* colc,
                                         v8f& a0, v8f& a1, v8f& a2, v8f& a3,
                                         int a_stride_y) {
  v2f a;
  a.x = ap[0];
  a.y = ap[a_stride_y];
  v2f b0, b1, b2, b3;
  b0.x = bp[colc[0]]; b0.y = bp[Ndim + colc[0]];
  b1.x = bp[colc[1]]; b1.y = bp[Ndim + colc[1]];
  b2.x = bp[colc[2]]; b2.y = bp[Ndim + colc[2]];
  b3.x = bp[colc[3]]; b3.y = bp[Ndim + colc[3]];
  a0 = wmma4(a, b0, a0);
  a1 = wmma4(a, b1, a1);
  a2 = wmma4(a, b2, a2);
  a3 = wmma4(a, b3, a3);
}

// ---------------------------------------------------------------------------
// Generic conv1x1:  Y[b][Cout,N] = W[Cout,Cin] * X[b][Cin,N] + bias
// block = 128 threads (4 waves); each wave owns a 16(M) x 64(N) output tile.
// ---------------------------------------------------------------------------
__global__ __launch_bounds__(128) void k_proj(
    const float* __restrict__ W, const float* __restrict__ bias,
    const float* __restrict__ X, float* __restrict__ Y,
    int Cout, int Cin)
{
  const int lane = threadIdx.x & 31;
  const int wv   = threadIdx.x >> 5;
  const int n0   = (blockIdx.x * 4 + wv) * 64;
  const int m0   = blockIdx.y * 16;
  const int b    = blockIdx.z;
  const float* Xb = X + (size_t)b * Cin * Ndim;
  float*       Yb = Y + (size_t)b * Cout * Ndim;

  const int half = lane >> 4;        // K pair {0,1} vs {2,3}
  const int l16  = lane & 15;
  int colc[4];
  #pragma unroll
  for (int t = 0; t < 4; ++t) {
    const int c = n0 + t * 16 + l16;
    colc[t] = c < Ndim ? c : 0;
  }

  const float* ap = W + (size_t)(m0 + l16) * Cin + half * 2;  // a.y at ap[1]
  const float* bp = Xb + (size_t)(half * 2) * Ndim;

  v8f acc0 = {}, acc1 = {}, acc2 = {}, acc3 = {};
  #pragma unroll 2
  for (int k0 = 0; k0 < Cin; k0 += 4) {
    mma_step(ap, bp, colc, acc0, acc1, acc2, acc3, /*a_stride_y=*/1);
    ap += 4;
    bp += 4 * Ndim;
  }

  v8f accs[4] = {acc0, acc1, acc2, acc3};
  #pragma unroll
  for (int t = 0; t < 4; ++t) {
    const int c = n0 + t * 16 + l16;
    if (c < Ndim) {
      #pragma unroll
      for (int r = 0; r < 8; ++r) {
        const int row = m0 + r + half * 8;
        Yb[(size_t)row * Ndim + c] = accs[t][r] + bias[row];
      }
    }
  }
}

// ---------------------------------------------------------------------------
// Wc1 over concat([desc, mh]) with fused eval-BN + ReLU.  Cin = Cout = 2C.
// K-loop split at the concat boundary (128 | 4) -> two clean pointer loops.
// ---------------------------------------------------------------------------
__global__ __launch_bounds__(128) void k_c1(
    const float* __restrict__ W, const float* __restrict__ g1,
    const float* __restrict__ b1, const float* __restrict__ X0,
    const float* __restrict__ X1, float* __restrict__ Y)
{
  constexpr int Cin = 2 * Cdim;
  const int lane = threadIdx.x & 31;
  const int wv   = threadIdx.x >> 5;
  const int n0   = (blockIdx.x * 4 + wv) * 64;
  const int m0   = blockIdx.y * 16;
  const int b    = blockIdx.z;
  const float* X0b = X0 + (size_t)b * Cdim * Ndim;
  const float* X1b = X1 + (size_t)b * Cdim * Ndim;
  float*       Yb  = Y  + (size_t)b * Cin * Ndim;

  const int half = lane >> 4;
  const int l16  = lane & 15;
  int colc[4];
  #pragma unroll
  for (int t = 0; t < 4; ++t) {
    const int c = n0 + t * 16 + l16;
    colc[t] = c < Ndim ? c : 0;
  }

  const float* ap = W + (size_t)(m0 + l16) * Cin + half * 2;
  v8f acc0 = {}, acc1 = {}, acc2 = {}, acc3 = {};

  const float* bp = X0b + (size_t)(half * 2) * Ndim;    // k in [0,128): desc
  #pragma unroll 2
  for (int k0 = 0; k0 < Cdim; k0 += 4) {
    mma_step(ap, bp, colc, acc0, acc1, acc2, acc3, 1);
    ap += 4;
    bp += 4 * Ndim;
  }
  bp = X1b + (size_t)(half * 2) * Ndim;                 // k in [128,256): mh
  #pragma unroll 2
  for (int k0 = 0; k0 < Cdim; k0 += 4) {
    mma_step(ap, bp, colc, acc0, acc1, acc2, acc3, 1);
    ap += 4;
    bp += 4 * Ndim;
  }

  const float bns = 0.9999950000374997f;  // 1/sqrt(1+1e-5)
  v8f accs[4] = {acc0, acc1, acc2, acc3};
  #pragma unroll
  for (int t = 0; t < 4; ++t) {
    const int c = n0 + t * 16 + l16;
    if (c < Ndim) {
      #pragma unroll
      for (int r = 0; r < 8; ++r) {
        const int row = m0 + r + half * 8;
        const float y = g1[row] * (accs[t][r] * bns) + b1[row];
        Yb[(size_t)row * Ndim + c] = fmaxf(y, 0.0f);
      }
    }
  }
}

// ---------------------------------------------------------------------------
// Wc2 with fused residual:  out = resid + Wc2*h + bc2.   Cin=2C, Cout=C.
// ---------------------------------------------------------------------------
__global__ __launch_bounds__(128) void k_c2(
    const float* __restrict__ W, const float* __restrict__ bias,
    const float* __restrict__ Xh, const float* __restrict__ resid,
    float* __restrict__ Y)
{
  constexpr int Cin = 2 * Cdim;
  const int lane = threadIdx.x & 31;
  const int wv   = threadIdx.x >> 5;
  const int n0   = (blockIdx.x * 4 + wv) * 64;
  const int m0   = blockIdx.y * 16;
  const int b    = blockIdx.z;
  const float* Xb = Xh    + (size_t)b * Cin * Ndim;
  const float* Rb = resid + (size_t)b * Cdim * Ndim;
  float*       Yb = Y     + (size_t)b * Cdim * Ndim;

  const int half = lane >> 4;
  const int l16  = lane & 15;
  int colc[4];
  #pragma unroll
  for (int t = 0; t < 4; ++t) {
    const int c = n0 + t * 16 + l16;
    colc[t] = c < Ndim ? c : 0;
  }

  const float* ap = W + (size_t)(m0 + l16) * Cin + half * 2;
  const float* bp = Xb + (size_t)(half * 2) * Ndim;
  v8f acc0 = {}, acc1 = {}, acc2 = {}, acc3 = {};
  #pragma unroll 2
  for (int k0 = 0; k0 < Cin; k0 += 4) {
    mma_step(ap, bp, colc, acc0, acc1, acc2, acc3, 1);
    ap += 4;
    bp += 4 * Ndim;
  }

  v8f accs[4] = {acc0, acc1, acc2, acc3};
  #pragma unroll
  for (int t = 0; t < 4; ++t) {
    const int c = n0 + t * 16 + l16;
    if (c < Ndim) {
      #pragma unroll
      for (int r = 0; r < 8; ++r) {
        const int row = m0 + r + half * 8;
        Yb[(size_t)row * Ndim + c] = Rb[(size_t)row * Ndim + c] + accs[t][r] + bias[row];
      }
    }
  }
}

// ---------------------------------------------------------------------------
// KNN-gather attention, flash-style online softmax.
// One thread per (b, h, n). q in 32 registers; key/val gathers use one base
// address per neighbor + constant immediate offsets (d*Ndim folded).
// ---------------------------------------------------------------------------
__global__ __launch_bounds__(256) void k_attn(
    const float* __restrict__ q, const float* __restrict__ key,
    const float* __restrict__ val, const int* __restrict__ neigh,
    const float* __restrict__ mask, const float* __restrict__ res_lam,
    float* __restrict__ add)
{
  const int n = blockIdx.x * blockDim.x + threadIdx.x;
  const int h = blockIdx.y;
  const int b = blockIdx.z;
  if (n >= Ndim) return;

  const float* qb = q   + ((size_t)b * Cdim + h * HD) * Ndim;
  const float* kb = key + ((size_t)b * Cdim + h * HD) * Ndim;
  const float* vb = val + ((size_t)b * Cdim + h * HD) * Ndim;
  const int*   nb = neigh + (size_t)b * Kn * Ndim + n;
  const float* mb = mask  + (size_t)b * Kn * Ndim + n;
  const float lam = res_lam[h];

  float qr[HD];
  #pragma unroll
  for (int d = 0; d < HD; ++d) qr[d] = qb[d * Ndim + n];

  const float scale = 0.17677669529663687f;  // 1/sqrt(32)
  float m = -INFINITY, l = 0.0f;
  float acc[HD];
  #pragma unroll
  for (int d = 0; d < HD; ++d) acc[d] = 0.0f;

  for (int k = 0; k < Kn; ++k) {
    const int idx = nb[k * Ndim];
    const float* kp = kb + idx;
    const float* vp = vb + idx;
    float s = 0.0f;
    #pragma unroll
    for (int d = 0; d < HD; ++d) s += qr[d] * kp[d * Ndim];
    s = s * scale + lam * mb[k * Ndim];
    s = fminf(fmaxf(s, -30.0f), 30.0f);
    const float mn   = fmaxf(m, s);
    const float cold = __expf(m - mn);   // exp(-inf)=0 on first iter
    const float p    = __expf(s - mn);
    l = l * cold + p;
    #pragma unroll
    for (int d = 0; d < HD; ++d) acc[d] = acc[d] * cold + p * vp[d * Ndim];
    m = mn;
  }
  // score = exp(s)/(sum_exp + 1e-8)  ->  acc / (l + 1e-8*exp(-m))
  const float inv = 1.0f / (l + 1e-8f * __expf(-m));
  float* ab = add + ((size_t)b * Cdim + h * HD) * Ndim;
  #pragma unroll
  for (int d = 0; d < HD; ++d) ab[d * Ndim + n] = acc[d] * inv;
}

// ---------------------------------------------------------------------------
// scores[b,n,m] = dot(p1[:,n], p2[:,m]) / sqrt(C), written into couplings.
// Each wave: 16 (rows) x 64 (cols); A reused across 4 back-to-back WMMAs.
// ---------------------------------------------------------------------------
__global__ __launch_bounds__(128) void k_scores(
    const float* __restrict__ p1, const float* __restrict__ p2,
    float* __restrict__ Cp)
{
  const int lane = threadIdx.x & 31;
  const int wv   = threadIdx.x >> 5;
  const int n0   = (blockIdx.x * 4 + wv) * 64;
  const int m0   = blockIdx.y * 16;
  const int b    = blockIdx.z;
  const float* p1b = p1 + (size_t)b * Cdim * Ndim;
  const float* p2b = p2 + (size_t)b * Cdim * Ndim;
  float*       Cb  = Cp + (size_t)b * NCP * NCP;

  const int half  = lane >> 4;
  const int l16   = lane & 15;
  const int arow  = m0 + l16;
  const int arowc = arow < Ndim ? arow : 0;
  int colc[4];
  #pragma unroll
  for (int t = 0; t < 4; ++t) {
    const int c = n0 + t * 16 + l16;
    colc[t] = c < Ndim ? c : 0;
  }

  // A is p1^T: a.x = p1[ka, row], a.y = p1[ka+1, row] (stride Ndim -> imm off)
  const float* ap = p1b + (size_t)(half * 2) * Ndim + arowc;
  const float* bp = p2b + (size_t)(half * 2) * Ndim;

  v8f acc0 = {}, acc1 = {}, acc2 = {}, acc3 = {};
  #pragma unroll 2
  for (int k0 = 0; k0 < Cdim; k0 += 4) {
    mma_step(ap, bp, colc, acc0, acc1, acc2, acc3, /*a_stride_y=*/Ndim);
    ap += 4 * Ndim;
    bp += 4 * Ndim;
  }

  const float rs = 0.08838834764831845f;  // 1/sqrt(128)
  v8f accs[4] = {acc0, acc1, acc2, acc3};
  #pragma unroll
  for (int t = 0; t < 4; ++t) {
    const int c = n0 + t * 16 + l16;
    if (c < Ndim) {
      #pragma unroll
      for (int r = 0; r < 8; ++r) {
        const int orow = m0 + r + half * 8;
        if (orow < Ndim) Cb[(size_t)orow * NCP + c] = accs[t][r] * rs;
      }
    }
  }
}

// Fill dustbin row/col with alpha; zero-init u and v.
__global__ void k_dustbin(float* __restrict__ Cp, const float* __restrict__ alpha_p,
                          float* __restrict__ u, float* __restrict__ v)
{
  const float alpha = *alpha_p;
  const int t = blockIdx.x * blockDim.x + threadIdx.x;
  const int b = blockIdx.y;
  float* Cb = Cp + (size_t)b * NCP * NCP;
  if (t < NCP) {
    Cb[(size_t)(NCP - 1) * NCP + t] = alpha;    // bottom row
    Cb[(size_t)t * NCP + (NCP - 1)] = alpha;    // right column
  }
  if (t < UVS) {
    u[(size_t)b * UVS + t] = 0.0f;
    v[(size_t)b * UVS + t] = 0.0f;
  }
}

// u_i = log_mu_i - LSE_j(Cp[b,i,j] + v_j).  One block per (i, b).
__global__ __launch_bounds__(256) void k_ulse(
    const float* __restrict__ Cp, const float* __restrict__ v,
    float* __restrict__ u, float norm, float logn)
{
  const int i = blockIdx.x, b = blockIdx.y, t = threadIdx.x;
  const float* row = Cp + ((size_t)b * NCP + i) * NCP;
  const float* vb  = v + (size_t)b * UVS;

  float m = -INFINITY, l = 0.0f;
  for (int j = t; j < NCP; j += 256) {
    const float x = row[j] + vb[j];
    if (x <= m) { l += __expf(x - m); }
    else        { l = l * __expf(m - x) + 1.0f; m = x; }
  }
  __shared__ float sm[256], sl[256];
  sm[t] = m; sl[t] = l;
  __syncthreads();
  for (int s = 128; s > 0; s >>= 1) {
    if (t < s) {
      const float m2 = sm[t + s], l2 = sl[t + s];
      const float mm = fmaxf(sm[t], m2);
      sl[t] = sl[t] * __expf(sm[t] - mm) + l2 * __expf(m2 - mm);
      sm[t] = mm;
    }
    __syncthreads();
  }
  if (t == 0) {
    const float lse = sm[0] + __logf(sl[0]);
    const float lmu = (i < NCP - 1) ? norm : (logn + norm);
    u[(size_t)b * UVS + i] = lmu - lse;
  }
}

// v_j = log_nu_j - LSE_i(Cp[b,i,j] + u_i).  Thread per column (coalesced rows).
__global__ __launch_bounds__(256) void k_vlse(
    const float* __restrict__ Cp, const float* __restrict__ u,
    float* __restrict__ v, float norm, float logn)
{
  const int j = blockIdx.x * blockDim.x + threadIdx.x;
  const int b = blockIdx.y;
  if (j >= NCP) return;
  const float* col = Cp + (size_t)b * NCP * NCP + j;
  const float* ub  = u + (size_t)b * UVS;

  float m = -INFINITY, l = 0.0f;
  for (int i = 0; i < NCP; ++i) {
    const float x = col[(size_t)i * NCP] + ub[i];
    if (x <= m) { l += __expf(x - m); }
    else        { l = l * __expf(m - x) + 1.0f; m = x; }
  }
  const float lnu = (j < NCP - 1) ? norm : (logn + norm);
  v[(size_t)b * UVS + j] = lnu - (m + __logf(l));
}

// out = exp(Cp + u_i + v_j - norm)
__global__ __launch_bounds__(256) void k_final(
    const float* __restrict__ Cp, const float* __restrict__ u,
    const float* __restrict__ v, float* __restrict__ out, float norm)
{
  const int j = blockIdx.x * blockDim.x + threadIdx.x;
  const int i = blockIdx.y, b = blockIdx.z;
  if (j >= NCP) return;
  const size_t idx = ((size_t)b * NCP + i) * NCP + j;
  out[idx] = __expf(Cp[idx] + u[(size_t)b * UVS + i] + v[(size_t)b * UVS + j] - norm);
}

}  // namespace

// ---------------------------------------------------------------------------
extern "C" void kernel_launch(void* const* d_in, const int* in_sizes, int n_in,
                              void* d_out, int out_size, void* d_ws, size_t ws_size,
                              hipStream_t stream) {
  (void)n_in; (void)out_size; (void)ws_size;

  const float* desc1  = (const float*)d_in[0];
  const float* desc2  = (const float*)d_in[1];
  const float* mask12 = (const float*)d_in[2];
  const float* mask21 = (const float*)d_in[3];

  // Detect flattening convention: insertion-order (params at 4, neigh at 21/22)
  // vs pytree-sorted (neigh at 4/5, params start at 6).
  const int* neigh12;
  const int* neigh21;
  int P;
  if (in_sizes[4] == Bdim * Kn * Ndim) {          // sorted top-level
    neigh12 = (const int*)d_in[4];
    neigh21 = (const int*)d_in[5];
    P = 6;
  } else {                                         // insertion order
    neigh12 = (const int*)d_in[21];
    neigh21 = (const int*)d_in[22];
    P = 4;
  }
  auto f = [&](int i) { return (const float*)d_in[i]; };
  const float *Wq, *bq, *Wk, *bk, *Wv, *bv, *Wmh, *bmh, *resl;
  const float *Wc1, *g1, *b1, *Wc2, *bc2, *Wf, *bf, *dust;
  if (in_sizes[P + 2] == (2 * Cdim) * (2 * Cdim)) {
    // params pytree-sorted: Wf, bf, {Wc1,Wc2,Wk,Wmh,Wq,Wv,b1,bc2,bk,bmh,bq,bv,g1,res_lam}, dustbin
    Wf  = f(P);      bf  = f(P + 1);
    Wc1 = f(P + 2);  Wc2 = f(P + 3);  Wk  = f(P + 4);  Wmh = f(P + 5);
    Wq  = f(P + 6);  Wv  = f(P + 7);  b1  = f(P + 8);  bc2 = f(P + 9);
    bk  = f(P + 10); bmh = f(P + 11); bq  = f(P + 12); bv  = f(P + 13);
    g1  = f(P + 14); resl = f(P + 15); dust = f(P + 16);
  } else {
    // insertion order: Wq,bq,Wk,bk,Wv,bv,Wmh,bmh,res_lam,Wc1,g1,b1,Wc2,bc2,Wf,bf,dustbin
    Wq  = f(P);      bq  = f(P + 1);  Wk  = f(P + 2);  bk  = f(P + 3);
    Wv  = f(P + 4);  bv  = f(P + 5);  Wmh = f(P + 6);  bmh = f(P + 7);
    resl = f(P + 8); Wc1 = f(P + 9);  g1  = f(P + 10); b1  = f(P + 11);
    Wc2 = f(P + 12); bc2 = f(P + 13); Wf  = f(P + 14); bf  = f(P + 15);
    dust = f(P + 16);
  }

  // Workspace carve (floats).
  float* ws = (float*)d_ws;
  size_t o = 0;
  auto carve = [&](size_t nelem) {
    float* p = ws + o;
    o += (nelem + 63) & ~(size_t)63;
    return p;
  };
  float* Cp = carve((size_t)Bdim * NCP * NCP);        // 72 MB couplings
  float* u  = carve((size_t)Bdim * UVS);
  float* v  = carve((size_t)Bdim * UVS);
  float* qb = carve((size_t)Bdim * Cdim * Ndim);
  float* kb = carve((size_t)Bdim * Cdim * Ndim);
  float* vb = carve((size_t)Bdim * Cdim * Ndim);
  float* ab = carve((size_t)Bdim * Cdim * Ndim);
  float* hb = carve((size_t)Bdim * 2 * Cdim * Ndim);
  float* d1 = carve((size_t)Bdim * Cdim * Ndim);
  float* d2 = carve((size_t)Bdim * Cdim * Ndim);

  const dim3 blk(128);
  const dim3 gP(12, Cdim / 16, Bdim);          // 12*4 waves * 64 cols = 3072
  const dim3 gC1(12, (2 * Cdim) / 16, Bdim);
  const dim3 gAttn((Ndim + 255) / 256, Hn, Bdim);

  auto attn_pass = [&](const float* dq, const float* dkv, const int* ng,
                       const float* mk, float* out) {
    k_proj<<<gP, blk, 0, stream>>>(Wq, bq, dq, qb, Cdim, Cdim);
    k_proj<<<gP, blk, 0, stream>>>(Wk, bk, dkv, kb, Cdim, Cdim);
    k_proj<<<gP, blk, 0, stream>>>(Wv, bv, dkv, vb, Cdim, Cdim);
    k_attn<<<gAttn, 256, 0, stream>>>(qb, kb, vb, ng, mk, resl, ab);
    k_proj<<<gP, blk, 0, stream>>>(Wmh, bmh, ab, qb, Cdim, Cdim);  // mh -> qb
    k_c1<<<gC1, blk, 0, stream>>>(Wc1, g1, b1, dq, qb, hb);
    k_c2<<<gP, blk, 0, stream>>>(Wc2, bc2, hb, dq, out);
  };
  attn_pass(desc1, desc2, neigh12, mask12, d1);
  attn_pass(desc2, desc1, neigh21, mask21, d2);

  // Final projections + big scores GEMM into couplings.
  k_proj<<<gP, blk, 0, stream>>>(Wf, bf, d1, qb, Cdim, Cdim);  // p1 -> qb
  k_proj<<<gP, blk, 0, stream>>>(Wf, bf, d2, kb, Cdim, Cdim);  // p2 -> kb
  k_scores<<<dim3(12, 188, Bdim), blk, 0, stream>>>(qb, kb, Cp);
  k_dustbin<<<dim3((UVS + 255) / 256, Bdim), 256, 0, stream>>>(Cp, dust, u, v);

  // Sinkhorn (couplings stays L2-resident: 72 MB < 192 MB L2).
  const float norm = -logf((float)(Ndim + Ndim));   // -log(m+n) = -log(6000)
  const float logn = logf((float)Ndim);
  for (int it = 0; it < SINK_ITERS; ++it) {
    k_ulse<<<dim3(NCP, Bdim), 256, 0, stream>>>(Cp, v, u, norm, logn);
    k_vlse<<<dim3((NCP + 255) / 256, Bdim), 256, 0, stream>>>(Cp, u, v, norm, logn);
  }
  k_final<<<dim3((NCP + 255) / 256, NCP, Bdim), 256, 0, stream>>>(
      Cp, u, v, (float*)d_out, norm);
}